// DecoderCategorical_63986422776329
// MI455X (gfx1250) — compile-verified
//
#include <hip/hip_runtime.h>

typedef float v2f __attribute__((ext_vector_type(2)));
typedef float v4f __attribute__((ext_vector_type(4)));
typedef float v8f __attribute__((ext_vector_type(8)));

#define DZ 64
#define DC 32
#define TILE 16
#define KB 16  /* 64 / 4 */

// D = A(beta tile, 16x4) x B(Z tile, 4x16) + C, chained over K=64.
// acc0 -> categories 0..15, acc1 -> categories 16..31.
// Lane L: holds timestep t0 + (L&15); component r of acc{0,1} is
// logit[t0+(L&15)][{0,16} + 8*(L>>4) + r].
__global__ __launch_bounds__(256)
void decoder_cat_wmma_kernel(const float* __restrict__ Z,
                             const float* __restrict__ beta,
                             float* __restrict__ out,
                             int num_tiles) {
  const int lane = threadIdx.x & 31;
  const int m    = lane & 15;   // row-in-tile / category-in-16
  const int hi   = lane >> 4;   // which K/M half this lane covers
  const int wave   = blockIdx.x * (blockDim.x >> 5) + (threadIdx.x >> 5);
  const int nwaves = gridDim.x * (blockDim.x >> 5);

  // ---- beta fragments (A matrix), loaded once per wave, L2-resident ----
  // A 16x4 f32 layout: lane L -> M = L%16, VGPR r -> K = 2*(L/16) + r.
  v2f bfrag[2][KB];
#pragma unroll
  for (int mt = 0; mt < 2; ++mt) {
    const float* brow = beta + (size_t)(m + 16 * mt) * DZ + 2 * hi;
#pragma unroll
    for (int kb = 0; kb < KB; ++kb)
      bfrag[mt][kb] = *(const v2f*)(brow + 4 * kb);
  }

  for (int tile = wave; tile < num_tiles; tile += nwaves) {
    const int t0 = tile * TILE;

    // ---- Z fragments (B matrix): lane L -> N = L%16, VGPR r -> K = 2*(L/16)+r
    const float* zrow = Z + (size_t)(t0 + m) * DZ + 2 * hi;
    v2f zfrag[KB];
#pragma unroll
    for (int kb = 0; kb < KB; ++kb)
      zfrag[kb] = __builtin_nontemporal_load((const v2f*)(zrow + 4 * kb));

    v8f acc0 = {0.f, 0.f, 0.f, 0.f, 0.f, 0.f, 0.f, 0.f};
    v8f acc1 = {0.f, 0.f, 0.f, 0.f, 0.f, 0.f, 0.f, 0.f};
#pragma unroll
    for (int kb = 0; kb < KB; ++kb) {
      acc0 = __builtin_amdgcn_wmma_f32_16x16x4_f32(
          false, bfrag[0][kb], false, zfrag[kb], (short)0, acc0, false, false);
      acc1 = __builtin_amdgcn_wmma_f32_16x16x4_f32(
          false, bfrag[1][kb], false, zfrag[kb], (short)0, acc1, false, false);
    }

    // ---- per-lane argmax over this lane's 16 categories (increasing index,
    // strict '>' keeps first occurrence; category 31 excluded -> -inf) ----
    float best = acc0[0];
    int   bidx = 8 * hi;
#pragma unroll
    for (int r = 1; r < 8; ++r) {
      float v = acc0[r];
      if (v > best) { best = v; bidx = 8 * hi + r; }
    }
#pragma unroll
    for (int r = 0; r < 8; ++r) {
      int c = 16 + 8 * hi + r;
      float v = (c == 31) ? -3.402823466e38f : acc1[r];
      if (v > best) { best = v; bidx = c; }
    }

    // ---- combine with partner lane (lane ^ 16) via ds_swizzle SWAPX16 ----
    int   pv_i = __builtin_amdgcn_ds_swizzle(__float_as_int(best), 0x401F);
    int   pidx = __builtin_amdgcn_ds_swizzle(bidx, 0x401F);
    float pv   = __int_as_float(pv_i);
    if (pv > best || (pv == best && pidx < bidx)) { best = pv; bidx = pidx; }

    // implicit last-category logit is 0; index 31 loses ties to earlier cats
    const int idx = (best >= 0.0f) ? bidx : 31;

    // ---- one-hot store: lane writes 64B half-row (coalesced, streaming) ----
    float* orow = out + (size_t)(t0 + m) * DC + 16 * hi;
#pragma unroll
    for (int j = 0; j < 4; ++j) {
      const int base = 16 * hi + 4 * j;
      v4f w;
      w.x = (idx == base + 0) ? 1.0f : 0.0f;
      w.y = (idx == base + 1) ? 1.0f : 0.0f;
      w.z = (idx == base + 2) ? 1.0f : 0.0f;
      w.w = (idx == base + 3) ? 1.0f : 0.0f;
      __builtin_nontemporal_store(w, (v4f*)(orow + 4 * j));
    }
  }
}

extern "C" void kernel_launch(void* const* d_in, const int* in_sizes, int n_in,
                              void* d_out, int out_size, void* d_ws, size_t ws_size,
                              hipStream_t stream) {
  const float* Z    = (const float*)d_in[0];   // (T, 64) fp32
  const float* beta = (const float*)d_in[1];   // (32, 64) fp32
  float* out = (float*)d_out;                  // (T, 32) fp32 one-hot

  const int T = in_sizes[0] / DZ;              // 2,000,000
  const int num_tiles = T / TILE;              // 125,000 (exact)

  const int threads = 256;                     // 8 waves / block
  int blocks = 1024;                           // grid-stride persistent-ish
  const int waves_needed = num_tiles;          // 1 tile per wave minimum
  if (blocks * (threads / 32) > waves_needed)
    blocks = (waves_needed + (threads / 32) - 1) / (threads / 32);
  if (blocks < 1) blocks = 1;

  decoder_cat_wmma_kernel<<<blocks, threads, 0, stream>>>(Z, beta, out, num_tiles);
}